// SDFRenderer_61667140436423
// MI455X (gfx1250) — compile-verified
//
#include <hip/hip_runtime.h>
#include <cstdint>

#define NSAMP 128
#define WAVES_PER_BLOCK 8
#define THREADS (WAVES_PER_BLOCK * 32)

typedef float v4f __attribute__((ext_vector_type(4)));

__global__ __launch_bounds__(THREADS) void sdf_render_kernel(
    const float* __restrict__ starts,
    const float* __restrict__ ends,
    const float* __restrict__ sdf,
    const int*   __restrict__ beta_p,
    float*       __restrict__ out,
    int n_rays)
{
    // one 2KB sdf slab per wave
    __shared__ __align__(16) float lds_sdf[WAVES_PER_BLOCK * NSAMP * 4];

    const int lane = threadIdx.x & 31;
    const int wv   = threadIdx.x >> 5;
    const int ray  = blockIdx.x * WAVES_PER_BLOCK + wv;
    if (ray >= n_rays) return;

    const float beta = (float)beta_p[0];

    float* wbuf = &lds_sdf[wv * (NSAMP * 4)];
    const float* gsdf = sdf + (size_t)ray * (NSAMP * 4);

    // ---- async-stage the 2KB sdf row into LDS: 4 x b128 per lane,
    //      each instruction is a fully contiguous 512B global read ----
    {
        // low 32 bits of a flat pointer into LDS == LDS byte offset
        unsigned lbase = (unsigned)(uintptr_t)(void*)wbuf + (unsigned)(lane * 16);
        unsigned long long gbase =
            (unsigned long long)(uintptr_t)(const void*)(gsdf + lane * 4);
        #pragma unroll
        for (int i = 0; i < 4; ++i) {
            unsigned           laddr = lbase + (unsigned)(i * 512);
            unsigned long long gaddr = gbase + (unsigned long long)(i * 512);
            asm volatile("global_load_async_to_lds_b128 %0, %1, off"
                         :: "v"(laddr), "v"(gaddr)
                         : "memory");
        }
        asm volatile("s_wait_asynccnt 0" ::: "memory");
    }

    // starts/ends: lane reads its 4 consecutive samples (contiguous 512B/wave).
    // One-touch data -> non-temporal loads (th:TH_LOAD_NT).
    const v4f s4 = __builtin_nontemporal_load(
        reinterpret_cast<const v4f*>(starts + (size_t)ray * NSAMP + lane * 4));
    const v4f e4 = __builtin_nontemporal_load(
        reinterpret_cast<const v4f*>(ends   + (size_t)ray * NSAMP + lane * 4));

    // lane's 4 samples of sdf (x,y,z = rgb, w = sdf_last), from LDS
    v4f sd[4];
    #pragma unroll
    for (int k = 0; k < 4; ++k)
        sd[k] = *reinterpret_cast<const v4f*>(wbuf + lane * 16 + k * 4);

    float pos[4] = { (s4.x + e4.x) * 0.5f, (s4.y + e4.y) * 0.5f,
                     (s4.z + e4.z) * 0.5f, (s4.w + e4.w) * 0.5f };

    float sl[4], al[4], u[4];
    #pragma unroll
    for (int k = 0; k < 4; ++k) {
        sl[k] = sd[k].w;
        // sigmoid(-sl*beta) = 1 / (1 + exp(sl*beta))
        float ex  = __expf(sl[k] * beta);
        float sig = 1.0f / (1.0f + ex);
        float e2  = __expf(-beta * sig);     // = 1 - alpha
        al[k] = 1.0f - e2;
        u[k]  = e2 + 1e-10f;                 // 1 - alpha + 1e-10 (no cancellation)
    }

    // per-lane cumulative products of the 4 local factors
    float c0 = u[0];
    float c1 = c0 * u[1];
    float c2 = c1 * u[2];
    float c3 = c2 * u[3];

    // wave-wide inclusive product scan of per-lane totals (Hillis-Steele)
    float incl = c3;
    #pragma unroll
    for (int d = 1; d < 32; d <<= 1) {
        float nn = __shfl_up(incl, d, 32);
        if (lane >= d) incl *= nn;
    }
    // exclusive prefix product for this lane's first sample
    float excl = __shfl_up(incl, 1, 32);
    if (lane == 0) excl = 1.0f;

    float t[4] = { excl, excl * c0, excl * c1, excl * c2 };

    float depth = 0.f, rr = 0.f, gg = 0.f, bb = 0.f;
    #pragma unroll
    for (int k = 0; k < 4; ++k) {
        float w = al[k] * t[k];
        depth += w * pos[k];
        rr    += w * sd[k].x;
        gg    += w * sd[k].y;
        bb    += w * sd[k].z;
    }

    // wave32 butterfly reduction
    #pragma unroll
    for (int d = 16; d > 0; d >>= 1) {
        depth += __shfl_xor(depth, d, 32);
        rr    += __shfl_xor(rr,    d, 32);
        gg    += __shfl_xor(gg,    d, 32);
        bb    += __shfl_xor(bb,    d, 32);
    }

    // output layout: [depth (N)] [rgb (N*3)] [sdf_last (N*128)] [positions (N*128)]
    float* out_depth = out;
    float* out_rgb   = out + n_rays;
    float* out_sl    = out + (size_t)n_rays * 4;
    float* out_pos   = out_sl + (size_t)n_rays * NSAMP;

    if (lane == 0) {
        out_depth[ray]       = depth;
        out_rgb[ray * 3 + 0] = rr;
        out_rgb[ray * 3 + 1] = gg;
        out_rgb[ray * 3 + 2] = bb;
    }
    // write-once streaming outputs -> non-temporal stores (th:TH_STORE_NT)
    v4f slv = { sl[0], sl[1], sl[2], sl[3] };
    v4f pv  = { pos[0], pos[1], pos[2], pos[3] };
    __builtin_nontemporal_store(
        slv, reinterpret_cast<v4f*>(out_sl  + (size_t)ray * NSAMP + lane * 4));
    __builtin_nontemporal_store(
        pv,  reinterpret_cast<v4f*>(out_pos + (size_t)ray * NSAMP + lane * 4));
}

extern "C" void kernel_launch(void* const* d_in, const int* in_sizes, int n_in,
                              void* d_out, int out_size, void* d_ws, size_t ws_size,
                              hipStream_t stream)
{
    const float* starts = (const float*)d_in[0];
    const float* ends   = (const float*)d_in[1];
    const float* sdf    = (const float*)d_in[2];
    const int*   beta   = (const int*)d_in[3];
    float* out = (float*)d_out;

    const int n_rays = in_sizes[0] / NSAMP;   // 65536
    const int blocks = (n_rays + WAVES_PER_BLOCK - 1) / WAVES_PER_BLOCK;
    sdf_render_kernel<<<blocks, THREADS, 0, stream>>>(starts, ends, sdf, beta, out, n_rays);
}